// FeatureEmbedding_24713241821784
// MI455X (gfx1250) — compile-verified
//
#include <hip/hip_runtime.h>
#include <hip/hip_bf16.h>

#define TT 1024
#define NN 16
#define DD 512
#define KH 8
#define EE 32768

typedef float v2f __attribute__((ext_vector_type(2)));
typedef float v8f __attribute__((ext_vector_type(8)));

// ---------------------------------------------------------------------------
// A: embed gather + PE add (store encPE) + GCN linear hW = enc @ W  [n,t,8]
// one wave32 per (n,t); lane handles 16 d-elements strided by 32
// ---------------------------------------------------------------------------
__global__ void k_embed_hw(const int* __restrict__ x, const float* __restrict__ PE,
                           const float* __restrict__ tab, const float* __restrict__ W,
                           float* __restrict__ encPE, float* __restrict__ hW) {
    int wid  = blockIdx.x * (blockDim.x >> 5) + (threadIdx.x >> 5);
    int lane = threadIdx.x & 31;
    int n = wid >> 10;
    int t = wid & (TT - 1);
    int id = x[t * NN + n];
    const float* erow  = tab + (size_t)id * DD;
    const float* perow = PE  + (size_t)t  * DD;
    float* eout = encPE + ((size_t)n * TT + t) * DD;
    float acc[KH];
#pragma unroll
    for (int k = 0; k < KH; ++k) acc[k] = 0.f;
#pragma unroll
    for (int c = 0; c < DD / 32; ++c) {
        int d = lane + c * 32;
        float e = erow[d];
        eout[d] = e + perow[d];
#pragma unroll
        for (int k = 0; k < KH; ++k) acc[k] += e * W[d * KH + k];
    }
#pragma unroll
    for (int off = 16; off >= 1; off >>= 1) {
#pragma unroll
        for (int k = 0; k < KH; ++k) acc[k] += __shfl_xor(acc[k], off, 32);
    }
    if (lane == 0) {
        float* ho = hW + ((size_t)n * TT + t) * KH;
#pragma unroll
        for (int k = 0; k < KH; ++k) ho[k] = acc[k];
    }
}

// ---------------------------------------------------------------------------
// B: degree / symmetric norm
// ---------------------------------------------------------------------------
__global__ void k_deg_init(float* __restrict__ deg) {
    int i = blockIdx.x * blockDim.x + threadIdx.x;
    if (i < TT) deg[i] = 1.0f;                 // self-loop
}
__global__ void k_deg_count(const int* __restrict__ ei, float* __restrict__ deg) {
    int e = blockIdx.x * blockDim.x + threadIdx.x;
    if (e < EE) atomicAdd(&deg[ei[EE + e]], 1.0f);   // col = edge_index[1]
}
__global__ void k_dis(const float* __restrict__ deg, float* __restrict__ dis) {
    int i = blockIdx.x * blockDim.x + threadIdx.x;
    if (i < TT) dis[i] = rsqrtf(deg[i]);
}

// ---------------------------------------------------------------------------
// C: gcn logits = bias + self-loop term, then edge scatter-add
// ---------------------------------------------------------------------------
__global__ void k_gcn_init(const float* __restrict__ hW, const float* __restrict__ dis,
                           const float* __restrict__ b, float* __restrict__ gcn) {
    int i = blockIdx.x * blockDim.x + threadIdx.x;     // [n][t][k] flat
    int k = i & (KH - 1);
    int t = (i >> 3) & (TT - 1);
    float dt = dis[t];
    gcn[i] = b[k] + hW[i] * dt * dt;
}
__global__ void k_gcn_edges(const int* __restrict__ ei, const float* __restrict__ hW,
                            const float* __restrict__ dis, float* __restrict__ gcn) {
    int i = blockIdx.x * blockDim.x + threadIdx.x;     // (n, e)
    int e = i & (EE - 1);
    int n = i >> 15;
    int r = ei[e];
    int c = ei[EE + e];
    float w = dis[r] * dis[c];
    const float* src = hW  + ((size_t)n * TT + r) * KH;
    float*       dst = gcn + ((size_t)n * TT + c) * KH;
#pragma unroll
    for (int k = 0; k < KH; ++k) atomicAdd(&dst[k], src[k] * w);
}

// ---------------------------------------------------------------------------
// D: softmax over t for att[n][k][t]; writes d_out att region and a
//    zero-padded [n][16][1024] copy so the WMMA kernel needs no predication
// ---------------------------------------------------------------------------
__global__ void k_softmax(const float* __restrict__ logits, float* __restrict__ attOut,
                          float* __restrict__ attPad) {
    int n = blockIdx.x >> 3;
    int k = blockIdx.x & 7;
    int tid = threadIdx.x;
    __shared__ float red[256];
    const float* src = logits + (size_t)n * TT * KH + k;    // stride KH over t
    float m = -3.4e38f;
    for (int t = tid; t < TT; t += 256) m = fmaxf(m, src[(size_t)t * KH]);
    red[tid] = m; __syncthreads();
    for (int s = 128; s > 0; s >>= 1) {
        if (tid < s) red[tid] = fmaxf(red[tid], red[tid + s]);
        __syncthreads();
    }
    float mx = red[0]; __syncthreads();
    float* dst = attOut + ((size_t)n * KH + k) * TT;
    float sum = 0.f;
    for (int t = tid; t < TT; t += 256) {
        float ev = __expf(src[(size_t)t * KH] - mx);
        dst[t] = ev;
        sum += ev;
    }
    red[tid] = sum; __syncthreads();
    for (int s = 128; s > 0; s >>= 1) {
        if (tid < s) red[tid] += red[tid + s];
        __syncthreads();
    }
    float inv = 1.0f / red[0];
    float* pad0 = attPad + ((size_t)n * 16 + k) * TT;
    float* pad1 = attPad + ((size_t)n * 16 + k + 8) * TT;
    for (int t = tid; t < TT; t += 256) {
        float v = dst[t] * inv;
        dst[t]  = v;
        pad0[t] = v;
        pad1[t] = 0.f;
    }
}

// ---------------------------------------------------------------------------
// E: learned_tokens[n,8,512] = att[n,16(pad),1024] @ encPE[n,1024,512]
//    one wave per (n, 16-wide column tile); 256 x V_WMMA_F32_16X16X4_F32
//    A frag: lanes 0-15 K={0,1}, lanes 16-31 K={2,3} (float2/lane)
//    B frag: same K split; N striped across lanes 0-15 / 16-31
//    D: VGPR r, lanes 0-15 -> row r (only rows 0..7 valid -> stored)
// ---------------------------------------------------------------------------
__global__ void k_wmma_tokens(const float* __restrict__ attPad,
                              const float* __restrict__ encPE,
                              float* __restrict__ out) {
    int wid  = blockIdx.x * (blockDim.x >> 5) + (threadIdx.x >> 5);
    int lane = threadIdx.x & 31;
    int n = wid >> 5;          // 16 batches x 32 column tiles
    int j = wid & 31;
    int m  = lane & 15;
    int hi = lane >> 4;
    const float* arow = attPad + ((size_t)n * 16 + m) * TT;
    const float* enc  = encPE  + (size_t)n * TT * DD;
    int col = j * 16 + m;
    v8f c = {};
    for (int kk = 0; kk < TT / 4; ++kk) {
        int kb = kk * 4 + hi * 2;
        v2f a = *(const v2f*)(arow + kb);
        v2f b;
        b.x = enc[(size_t)kb * DD + col];
        b.y = enc[(size_t)(kb + 1) * DD + col];
        c = __builtin_amdgcn_wmma_f32_16x16x4_f32(
                /*neg_a=*/false, a, /*neg_b=*/false, b,
                /*c_mod=*/(short)0, c, /*reuse_a=*/false, /*reuse_b=*/false);
    }
    if (hi == 0) {
        float* op = out + ((size_t)n * 9 + 1) * DD + col;   // rows 1..8 of out
#pragma unroll
        for (int r = 0; r < 8; ++r) op[(size_t)r * DD] = c[r];
    }
}

// ---------------------------------------------------------------------------
// F: code token -> out[n][0][:]
// ---------------------------------------------------------------------------
__global__ void k_code(const float* __restrict__ ct, float* __restrict__ out) {
    int i = blockIdx.x * blockDim.x + threadIdx.x;   // [n][d]
    int n = i >> 9;
    int d = i & (DD - 1);
    out[(size_t)n * 9 * DD + d] = ct[d];
}

// ---------------------------------------------------------------------------
extern "C" void kernel_launch(void* const* d_in, const int* in_sizes, int n_in,
                              void* d_out, int out_size, void* d_ws, size_t ws_size,
                              hipStream_t stream) {
    (void)in_sizes; (void)n_in; (void)out_size; (void)ws_size;
    const int*   x   = (const int*)d_in[0];
    const int*   ei  = (const int*)d_in[1];
    const float* PE  = (const float*)d_in[2];
    const float* tab = (const float*)d_in[3];
    const float* ct  = (const float*)d_in[4];
    const float* W   = (const float*)d_in[5];
    const float* b   = (const float*)d_in[6];
    float* out = (float*)d_out;
    float* ws  = (float*)d_ws;

    // workspace layout (floats)
    float* encPE  = ws;                                   // 16*1024*512
    float* hW     = encPE  + (size_t)NN * TT * DD;        // 16*1024*8
    float* gcn    = hW     + (size_t)NN * TT * KH;        // 16*1024*8
    float* deg    = gcn    + (size_t)NN * TT * KH;        // 1024
    float* dis    = deg    + TT;                          // 1024
    float* attPad = dis    + TT;                          // 16*16*1024
    float* attOut = out    + (size_t)NN * 9 * DD;         // second output (att)

    k_embed_hw  <<<NN * TT / 8,        256, 0, stream>>>(x, PE, tab, W, encPE, hW);
    k_deg_init  <<<TT / 256,           256, 0, stream>>>(deg);
    k_deg_count <<<EE / 256,           256, 0, stream>>>(ei, deg);
    k_dis       <<<TT / 256,           256, 0, stream>>>(deg, dis);
    k_gcn_init  <<<NN * TT * KH / 256, 256, 0, stream>>>(hW, dis, b, gcn);
    k_gcn_edges <<<NN * EE / 256,      256, 0, stream>>>(ei, hW, dis, gcn);
    k_softmax   <<<NN * KH,            256, 0, stream>>>(gcn, attOut, attPad);
    k_code      <<<NN * DD / 256,      256, 0, stream>>>(ct, out);
    k_wmma_tokens<<<NN * 32 / 8,       256, 0, stream>>>(attPad, encPE, out);
}